// SemiHardTripletLoss_13546326852330
// MI455X (gfx1250) — compile-verified
//
#include <hip/hip_runtime.h>
#include <hip/hip_bf16.h>

#define N_ROWS 8192
#define DIM    512
#define NCHUNK 8
#define CHUNK  1024            // columns per block (8192 / NCHUNK)
#define MARGIN_F 0.35f

typedef float v2f __attribute__((ext_vector_type(2)));
typedef float v8f __attribute__((ext_vector_type(8)));

// ---------------------------------------------------------------------------
// Kernel 1: L2-normalize each row. One wave (32 lanes) per row, 8 rows/block.
// ---------------------------------------------------------------------------
__global__ __launch_bounds__(256) void norm_kernel(const float* __restrict__ feat,
                                                   float* __restrict__ emb) {
    const int tid  = threadIdx.x;
    const int wave = tid >> 5;
    const int lane = tid & 31;
    const int row  = blockIdx.x * 8 + wave;

    const float4* src = (const float4*)(feat + (size_t)row * DIM);
    float4*       dst = (float4*)(emb + (size_t)row * DIM);

    float4 v[4];
    float ss = 0.0f;
#pragma unroll
    for (int j = 0; j < 4; ++j) {
        v[j] = src[lane + j * 32];
        ss += v[j].x * v[j].x + v[j].y * v[j].y + v[j].z * v[j].z + v[j].w * v[j].w;
    }
#pragma unroll
    for (int off = 16; off >= 1; off >>= 1)
        ss += __shfl_xor(ss, off, 32);

    const float inv = 1.0f / sqrtf(ss);
#pragma unroll
    for (int j = 0; j < 4; ++j) {
        float4 o;
        o.x = v[j].x * inv; o.y = v[j].y * inv;
        o.z = v[j].z * inv; o.w = v[j].w * inv;
        dst[lane + j * 32] = o;
    }
}

// ---------------------------------------------------------------------------
// Kernel 2: fused Gram-matrix tiles (f32 WMMA 16x16x4) + masked row min/max.
// grid = (N/16 row tiles, NCHUNK column chunks), block = 256 (8 waves).
// Wave w owns 8 16x16 C tiles; processed as 2 passes of 4 concurrent
// accumulators so each A fragment feeds 4 WMMAs, with register
// double-buffering so next-step loads overlap current-step WMMAs.
// A and B fragments use the identical row-major 16x4 f32 layout:
//   lane l: row (l&15), K elements 2*(l>>4) and 2*(l>>4)+1.
// ---------------------------------------------------------------------------
__global__ __launch_bounds__(256) void sims_kernel(const float* __restrict__ emb,
                                                   const int* __restrict__ labels,
                                                   float* __restrict__ posPart,
                                                   float* __restrict__ negPart) {
    __shared__ int   colLabLds[CHUNK];
    __shared__ int   rowLabLds[16];
    __shared__ float ldsPos[8][16];
    __shared__ float ldsNeg[8][16];

    const int tid  = threadIdx.x;
    const int wave = tid >> 5;
    const int lane = tid & 31;
    const int half = lane >> 4;     // 0: rows m=0..7 / K 0,1   1: rows m=8..15 / K 2,3
    const int l15  = lane & 15;

    const int rowBase   = blockIdx.x * 16;
    const int chunkBase = blockIdx.y * CHUNK;

    for (int i = tid; i < CHUNK; i += 256) colLabLds[i] = labels[chunkBase + i];
    if (tid < 16) rowLabLds[tid] = labels[rowBase + tid];
    __syncthreads();

    int rowLab[8];
#pragma unroll
    for (int r = 0; r < 8; ++r) rowLab[r] = rowLabLds[r + 8 * half];

    float minp[8], maxn[8];
#pragma unroll
    for (int r = 0; r < 8; ++r) { minp[r] = 1.0f; maxn[r] = -1.0f; }

    // A-fragment base: row (rowBase + l15), K offset 2*half
    const float* aBase = emb + (size_t)(rowBase + l15) * DIM + 2 * half;

    for (int tp = 0; tp < 2; ++tp) {           // 2 passes x 4 tiles = 8 tiles
        const int tileBase = tp * 4;

        const float* bBase[4];
#pragma unroll
        for (int t = 0; t < 4; ++t) {
            const int colBase = chunkBase + wave * 16 + (tileBase + t) * 128;
            bBase[t] = emb + (size_t)(colBase + l15) * DIM + 2 * half;
        }

        v8f acc[4];
#pragma unroll
        for (int t = 0; t < 4; ++t) acc[t] = (v8f){};

        // prologue loads (K step 0)
        v2f aCur = *(const v2f*)(aBase);
        v2f bCur[4];
#pragma unroll
        for (int t = 0; t < 4; ++t) bCur[t] = *(const v2f*)(bBase[t]);

#pragma unroll 4
        for (int k = 0; k < DIM; k += 4) {
            // issue next step's loads before consuming current fragments
            const int kn = (k + 4 < DIM) ? (k + 4) : 0;   // uniform, branchless wrap
            v2f aNxt = *(const v2f*)(aBase + kn);
            v2f bNxt[4];
#pragma unroll
            for (int t = 0; t < 4; ++t) bNxt[t] = *(const v2f*)(bBase[t] + kn);

            // 4 independent accumulator chains share one A fragment
            acc[0] = __builtin_amdgcn_wmma_f32_16x16x4_f32(
                false, aCur, false, bCur[0], (short)0, acc[0], false, false);
            acc[1] = __builtin_amdgcn_wmma_f32_16x16x4_f32(
                false, aCur, false, bCur[1], (short)0, acc[1], false, false);
            acc[2] = __builtin_amdgcn_wmma_f32_16x16x4_f32(
                false, aCur, false, bCur[2], (short)0, acc[2], false, false);
            acc[3] = __builtin_amdgcn_wmma_f32_16x16x4_f32(
                false, aCur, false, bCur[3], (short)0, acc[3], false, false);

            aCur = aNxt;
#pragma unroll
            for (int t = 0; t < 4; ++t) bCur[t] = bNxt[t];
        }

        // masked min/max for the 4 finished tiles
#pragma unroll
        for (int t = 0; t < 4; ++t) {
            const int tileOff = wave * 16 + (tileBase + t) * 128;
            const int colBase = chunkBase + tileOff;
            const int colLab  = colLabLds[tileOff + l15];
            const int gc      = colBase + l15;
#pragma unroll
            for (int r = 0; r < 8; ++r) {
                float c = fminf(fmaxf(acc[t][r], -1.0f), 1.0f);  // clip like reference
                const int  gr   = rowBase + r + 8 * half;
                const bool same = (rowLab[r] == colLab);
                const float pv  = (same && (gr != gc)) ? c : 1.0f;  // pos fill 1.0
                const float nv  = same ? -1.0f : c;                 // neg fill -1.0
                minp[r] = fminf(minp[r], pv);
                maxn[r] = fmaxf(maxn[r], nv);
            }
        }
    }

    // Reduce across the 16 lanes of each half (they share the same 8 rows).
#pragma unroll
    for (int r = 0; r < 8; ++r) {
        float p = minp[r], n = maxn[r];
#pragma unroll
        for (int off = 8; off >= 1; off >>= 1) {
            p = fminf(p, __shfl_xor(p, off, 32));
            n = fmaxf(n, __shfl_xor(n, off, 32));
        }
        minp[r] = p; maxn[r] = n;
    }
    if (l15 == 0) {
#pragma unroll
        for (int r = 0; r < 8; ++r) {
            ldsPos[wave][r + 8 * half] = minp[r];
            ldsNeg[wave][r + 8 * half] = maxn[r];
        }
    }
    __syncthreads();

    if (tid < 16) {
        float p = 1.0f, n = -1.0f;
#pragma unroll
        for (int w = 0; w < 8; ++w) {
            p = fminf(p, ldsPos[w][tid]);
            n = fmaxf(n, ldsNeg[w][tid]);
        }
        posPart[(size_t)blockIdx.y * N_ROWS + rowBase + tid] = p;
        negPart[(size_t)blockIdx.y * N_ROWS + rowBase + tid] = n;
    }
}

// ---------------------------------------------------------------------------
// Kernel 3: reduce chunk partials -> per-row loss -> scalar mean over valid.
// ---------------------------------------------------------------------------
__global__ __launch_bounds__(256) void final_kernel(const float* __restrict__ posPart,
                                                    const float* __restrict__ negPart,
                                                    float* __restrict__ out) {
    __shared__ float sSum[256];
    __shared__ int   sCnt[256];

    float sum = 0.0f;
    int   cnt = 0;
    for (int i = threadIdx.x; i < N_ROWS; i += 256) {
        float p = 1.0f, n = -1.0f;
#pragma unroll
        for (int ch = 0; ch < NCHUNK; ++ch) {
            p = fminf(p, posPart[(size_t)ch * N_ROWS + i]);
            n = fmaxf(n, negPart[(size_t)ch * N_ROWS + i]);
        }
        float l = fmaxf(MARGIN_F + n - p, 0.0f);
        sum += l;
        cnt += (l > 0.0f) ? 1 : 0;
    }
    sSum[threadIdx.x] = sum;
    sCnt[threadIdx.x] = cnt;
    __syncthreads();
    for (int s = 128; s > 0; s >>= 1) {
        if (threadIdx.x < s) {
            sSum[threadIdx.x] += sSum[threadIdx.x + s];
            sCnt[threadIdx.x] += sCnt[threadIdx.x + s];
        }
        __syncthreads();
    }
    if (threadIdx.x == 0) {
        const float fl = sSum[0];
        const float nv = (float)sCnt[0];
        out[0] = (nv > 0.0f) ? fl / fmaxf(nv, 1.0f) : fl;
    }
}

// ---------------------------------------------------------------------------
extern "C" void kernel_launch(void* const* d_in, const int* in_sizes, int n_in,
                              void* d_out, int out_size, void* d_ws, size_t ws_size,
                              hipStream_t stream) {
    const float* feat   = (const float*)d_in[0];
    const int*   labels = (const int*)d_in[1];
    float*       out    = (float*)d_out;

    char*  ws      = (char*)d_ws;
    float* emb     = (float*)ws;                                          // 16 MB
    float* posPart = (float*)(ws + (size_t)N_ROWS * DIM * sizeof(float)); // 256 KB
    float* negPart = posPart + (size_t)NCHUNK * N_ROWS;                   // 256 KB

    norm_kernel<<<N_ROWS / 8, 256, 0, stream>>>(feat, emb);

    dim3 grid(N_ROWS / 16, NCHUNK);
    sims_kernel<<<grid, 256, 0, stream>>>(emb, labels, posPart, negPart);

    final_kernel<<<1, 256, 0, stream>>>(posPart, negPart, out);
}